// YoloLayer2_80659485819209
// MI455X (gfx1250) — compile-verified
//
#include <hip/hip_runtime.h>
#include <math.h>

// Problem constants (from reference)
#define B_      16
#define T_      50
#define A_ANCH  3
#define NCLS    80
#define H_      76
#define W_      76
#define HW      (H_*W_)             // 5776
#define CH      (5+NCLS)            // 85
#define NRUNS   (B_*A_ANCH)         // 48 conf runs of HW floats
#define SEGS    2
#define NBLK_CONF (NRUNS*SEGS)      // 96
#define NT      (B_*T_)             // 800 targets
#define NCELLS  (B_*A_ANCH*HW)      // 277248
#define NMASKW  ((NCELLS+31)/32)    // 8664 bitmask words
#define TPB     256

// workspace layout (float-word offsets)
#define OFF_CONF   0                 // [96]  per-block conf partials
#define OFF_COORD  256               // [800] per-target coord loss
#define OFF_CLS    1056              // [800] per-target cls loss
#define OFF_VALID  1856              // [800] per-target valid flag
#define OFF_MASK   2656              // [8664] uint32 assigned-cell bitmask

typedef float v2f_t __attribute__((ext_vector_type(2)));
typedef float v8f_t __attribute__((ext_vector_type(8)));

__device__ __forceinline__ float sigmoidf_(float z) { return 1.0f / (1.0f + expf(-z)); }
// softplus(z) = log(1+e^z); equals -log(sigmoid(-z)). Stable BCE building block.
__device__ __forceinline__ float softplusf_(float z) { return (z > 15.0f) ? z : log1pf(expf(z)); }

__global__ void k_init(unsigned int* __restrict__ mask) {
    int i = blockIdx.x * blockDim.x + threadIdx.x;
    if (i < NMASKW) mask[i] = 0u;
}

// One thread per target: anchor argmax, gathers, coord L1 + class BCE, mark assigned cell.
__global__ void k_targets(const float* __restrict__ inp, const float* __restrict__ tgt,
                          float* __restrict__ tCoord, float* __restrict__ tCls,
                          float* __restrict__ tValid, unsigned int* __restrict__ mask) {
    int idx = blockIdx.x * blockDim.x + threadIdx.x;
    if (idx >= NT) return;
    int b = idx / T_, t = idx % T_;
    const float* tg = tgt + (size_t)b * (T_ * 5) + t * 5;
    float cls = tg[0], x = tg[1], y = tg[2], w = tg[3], h = tg[4];
    bool valid = ((((cls + x) + y) + w) + h) != 0.0f;

    const float ANC[9][2] = {{10.f,13.f},{16.f,30.f},{33.f,23.f},{30.f,61.f},{62.f,45.f},
                             {59.f,119.f},{116.f,90.f},{156.f,198.f},{373.f,326.f}};
    float bestR = -1.0f; int best = 0;
    #pragma unroll
    for (int k = 0; k < 9; k++) {
        float aw = ANC[k][0] * (1.0f/608.0f), ah = ANC[k][1] * (1.0f/608.0f);
        float inter = fminf(w, aw) * fminf(h, ah);
        float uni   = w*h + aw*ah - inter;
        float r = inter / uni;
        if (r > bestR) { bestR = r; best = k; }   // strict > keeps first max (jnp.argmax)
    }
    bool assigned = valid && (best < A_ANCH);

    float lc = 0.0f, lcls = 0.0f;
    if (assigned) {
        int gi = (int)(x * (float)W_); gi = gi < 0 ? 0 : (gi > W_-1 ? W_-1 : gi);
        int gj = (int)(y * (float)H_); gj = gj < 0 ? 0 : (gj > H_-1 ? H_-1 : gj);
        int cell = ((b * A_ANCH + best) * H_ + gj) * W_ + gi;
        atomicOr(&mask[cell >> 5], 1u << (cell & 31));   // idempotent -> deterministic

        size_t base = ((size_t)b * (A_ANCH*CH) + (size_t)best * CH) * HW + (size_t)gj * W_ + gi;
        float px = sigmoidf_(inp[base]);
        float py = sigmoidf_(inp[base + (size_t)HW]);
        float pw = inp[base + 2*(size_t)HW];
        float ph = inp[base + 3*(size_t)HW];
        float aw = ANC[best][0], ah = ANC[best][1];
        float txt = x * (float)W_ - (float)gi;
        float tyt = y * (float)H_ - (float)gj;
        float twt = logf(w * 608.0f / aw);
        float tht = logf(h * 608.0f / ah);
        float scale = 2.0f * y * w;   // faithful: 2 * t[2] * t[3]
        lc = scale * (fabsf(px - txt) + fabsf(py - tyt) + fabsf(pw - twt) + fabsf(ph - tht));

        int tc = (int)cls;
        const float* pz = inp + base + 5*(size_t)HW;
        for (int c = 0; c < NCLS; c++) {
            float z = pz[(size_t)c * HW];
            lcls += softplusf_(z);        // BCE = softplus(z) - onehot*z
            if (c == tc) lcls -= z;
        }
    }
    tCoord[idx] = lc;
    tCls[idx]   = lcls;
    tValid[idx] = valid ? 1.0f : 0.0f;
}

// Sum sigmoid(conf) over a half-run of contiguous floats; wave reduction via V_WMMA_F32_16X16X4_F32.
__global__ void k_confsum(const float* __restrict__ inp, float* __restrict__ confPartial) {
    int bi  = blockIdx.x;
    int run = bi >> 1, seg = bi & 1;
    int b = run / A_ANCH, a = run % A_ANCH;
    const float4* p = (const float4*)(inp + ((size_t)b * (A_ANCH*CH) + (size_t)a * CH + 4) * HW);
    const int nvec = HW / 4;            // 1444, runs are 16B aligned
    int beg = seg * (nvec / 2);         // 722 float4 per segment
    int end = beg + (nvec / 2);

    float sloc = 0.0f;
    for (int i = beg + (int)threadIdx.x; i < end; i += TPB) {
        float4 v = p[i];
        sloc += sigmoidf_(v.x) + sigmoidf_(v.y) + sigmoidf_(v.z) + sigmoidf_(v.w);
    }

    // Wave32 sum on the matrix pipe: A(16x4 f32) holds lane partials, B = ones(4x16).
    // D[m][n] = sum_k A[m][k]; lanes 0-15 hold rows 0-7, lanes 16-31 hold rows 8-15.
    v2f_t av; av[0] = sloc; av[1] = 0.0f;
    v2f_t bv; bv[0] = 1.0f; bv[1] = 1.0f;
    v8f_t cz = {0.f,0.f,0.f,0.f,0.f,0.f,0.f,0.f};
    v8f_t d = __builtin_amdgcn_wmma_f32_16x16x4_f32(
        false, av, false, bv, (short)0, cz, false, false);
    float s = d[0]+d[1]+d[2]+d[3]+d[4]+d[5]+d[6]+d[7];
    float wsum = __shfl(s, 0, 32) + __shfl(s, 16, 32);   // rows 0-7 + rows 8-15

    __shared__ float sh[TPB/32];
    int lane = threadIdx.x & 31, wid = threadIdx.x >> 5;
    if (lane == 0) sh[wid] = wsum;
    __syncthreads();
    if (threadIdx.x == 0) {
        float tot = 0.0f;
        for (int i = 0; i < TPB/32; i++) tot += sh[i];   // fixed order -> deterministic
        confPartial[bi] = tot;
    }
}

__device__ __forceinline__ float blockReduce(float v, float* sh) {
    int tid = threadIdx.x;
    sh[tid] = v; __syncthreads();
    for (int s = TPB/2; s > 0; s >>= 1) {
        if (tid < s) sh[tid] += sh[tid + s];
        __syncthreads();
    }
    float r = sh[0]; __syncthreads();
    return r;
}

__global__ void k_finalize(const float* __restrict__ confPartial, const float* __restrict__ tCoord,
                           const float* __restrict__ tCls, const float* __restrict__ tValid,
                           const unsigned int* __restrict__ mask, float* __restrict__ out) {
    __shared__ float sh[TPB];
    int tid = threadIdx.x;
    float sConf = 0.f, sCoord = 0.f, sCls = 0.f, sValid = 0.f, sDist = 0.f;
    for (int i = tid; i < NBLK_CONF; i += TPB) sConf += confPartial[i];
    for (int i = tid; i < NT; i += TPB) { sCoord += tCoord[i]; sCls += tCls[i]; sValid += tValid[i]; }
    for (int i = tid; i < NMASKW; i += TPB) sDist += (float)__popc(mask[i]);

    sConf  = blockReduce(sConf,  sh);
    sCoord = blockReduce(sCoord, sh);
    sCls   = blockReduce(sCls,   sh);
    sValid = blockReduce(sValid, sh);
    sDist  = blockReduce(sDist,  sh);

    if (tid == 0) {
        float nt = sValid > 1.0f ? sValid : 1.0f;
        // -sum(loss_obj) = sum(sigmoid(conf)) - #distinct assigned cells
        out[0] = (sConf - sDist) + sCoord / nt + sCls;
    }
}

extern "C" void kernel_launch(void* const* d_in, const int* in_sizes, int n_in,
                              void* d_out, int out_size, void* d_ws, size_t ws_size,
                              hipStream_t stream) {
    const float* inp = (const float*)d_in[0];
    const float* tgt = (const float*)d_in[1];
    float* ws = (float*)d_ws;
    float*        confPartial = ws + OFF_CONF;
    float*        tCoord      = ws + OFF_COORD;
    float*        tCls        = ws + OFF_CLS;
    float*        tValid      = ws + OFF_VALID;
    unsigned int* mask        = (unsigned int*)(ws + OFF_MASK);

    k_init   <<<(NMASKW + TPB - 1)/TPB, TPB, 0, stream>>>(mask);
    k_targets<<<(NT + TPB - 1)/TPB,     TPB, 0, stream>>>(inp, tgt, tCoord, tCls, tValid, mask);
    k_confsum<<<NBLK_CONF,              TPB, 0, stream>>>(inp, confPartial);
    k_finalize<<<1,                     TPB, 0, stream>>>(confPartial, tCoord, tCls, tValid, mask,
                                                          (float*)d_out);
}